// GarenBCPolicy_32658931319072
// MI455X (gfx1250) — compile-verified
//
#include <hip/hip_runtime.h>
#include <hip/hip_bf16.h>

typedef float v2f __attribute__((ext_vector_type(2)));
typedef float v8f __attribute__((ext_vector_type(8)));

#define TT 66
#define OBS 21408

// ---------------------------------------------------------------------------
// Precompute 1: base_h_{s,m}[t][c] = char_emb[t,:] @ W1[0:128,:] + b1   (pre-ReLU)
// ---------------------------------------------------------------------------
__global__ __launch_bounds__(256) void garen_base_kernel(
    const float* __restrict__ char_emb,
    const float* __restrict__ Ws1, const float* __restrict__ bs1,
    const float* __restrict__ Wm1, const float* __restrict__ bm1,
    float* __restrict__ base_h_s, float* __restrict__ base_h_m)
{
    int idx = blockIdx.x * 256 + threadIdx.x;
    if (idx < TT * 128) {
        int t = idx >> 7, c = idx & 127;
        float s = bs1[c];
        const float* ce = char_emb + t * 128;
        #pragma unroll 8
        for (int j = 0; j < 128; ++j) s = fmaf(ce[j], Ws1[j * 128 + c], s);
        base_h_s[idx] = s;
    } else if (idx < 2 * TT * 128) {
        int i2 = idx - TT * 128;
        int t = i2 >> 7, c = i2 & 127;
        float s = bm1[c];
        const float* ce = char_emb + t * 128;
        #pragma unroll 8
        for (int j = 0; j < 128; ++j) s = fmaf(ce[j], Wm1[j * 128 + c], s);
        base_h_m[i2] = s;
    }
}

// ---------------------------------------------------------------------------
// Precompute 2: def_out_{s,m}[t][c] = relu(base_h[t,:]) @ W2 + b2
// (the output value for every char slot that received no detection)
// ---------------------------------------------------------------------------
__global__ __launch_bounds__(256) void garen_def_kernel(
    const float* __restrict__ base_h_s, const float* __restrict__ base_h_m,
    const float* __restrict__ Ws2, const float* __restrict__ bs2,
    const float* __restrict__ Wm2, const float* __restrict__ bm2,
    float* __restrict__ def_out_s, float* __restrict__ def_out_m)
{
    int idx = blockIdx.x * 256 + threadIdx.x;
    if (idx < TT * 128) {
        int t = idx >> 7, c = idx & 127;
        float s = bs2[c];
        const float* h = base_h_s + t * 128;
        #pragma unroll 8
        for (int j = 0; j < 128; ++j) s = fmaf(fmaxf(h[j], 0.f), Ws2[j * 128 + c], s);
        def_out_s[idx] = s;
    } else if (idx < 2 * TT * 128) {
        int i2 = idx - TT * 128;
        int t = i2 >> 7, c = i2 & 127;
        float s = bm2[c];
        const float* h = base_h_m + t * 128;
        #pragma unroll 8
        for (int j = 0; j < 128; ++j) s = fmaf(fmaxf(h[j], 0.f), Wm2[j * 128 + c], s);
        def_out_m[i2] = s;
    }
}

// ---------------------------------------------------------------------------
// Main kernel: one block (8 wave32) per batch row.
//   phase 1: build active-row layer-1 activations h into LDS (rank-4 / rank-2
//            update on precomputed base_h, then ReLU)
//   phase 2: layer-2 GEMM via v_wmma_f32_16x16x4_f32, scatter-store to out
//   phase 3: default-row copies, item-embedding gather, continuous prefix
// ---------------------------------------------------------------------------
__global__ __launch_bounds__(256) void garen_main_kernel(
    const float* __restrict__ cont,
    const float* __restrict__ sfeat,      // [B,24,4]
    const float* __restrict__ mfeat,      // [B,40,2]
    const float* __restrict__ item_emb,   // [287,64]
    const float* __restrict__ Ws1,        // [132,128]
    const float* __restrict__ Ws2,        // [128,128]
    const float* __restrict__ bs2,
    const float* __restrict__ Wm1,        // [130,128]
    const float* __restrict__ Wm2,
    const float* __restrict__ bm2,
    const int*   __restrict__ sids,       // [B,24]
    const int*   __restrict__ mids,       // [B,40]
    const int*   __restrict__ items,      // [B,70]
    const float* __restrict__ base_h_s,
    const float* __restrict__ base_h_m,
    const float* __restrict__ def_out_s,
    const float* __restrict__ def_out_m,
    float* __restrict__ out)
{
    // row stride 132 floats -> bank = (4*row + k) mod 64, conflict-free b64 loads
    __shared__ float h_s[32 * 132];
    __shared__ float h_m[48 * 132];
    __shared__ int   s_ids[24];
    __shared__ int   m_ids[40];
    __shared__ int   act_s[TT];
    __shared__ int   act_m[TT];

    const int b   = blockIdx.x;
    const int tid = threadIdx.x;
    const size_t obase = (size_t)b * OBS;

    if (tid < TT) { act_s[tid] = 0; act_m[tid] = 0; }
    if (tid < 24) s_ids[tid] = sids[b * 24 + tid];
    if (tid < 40) m_ids[tid] = mids[b * 40 + tid];
    __syncthreads();
    if (tid < 24) act_s[s_ids[tid]] = 1;
    if (tid < 40) act_m[m_ids[tid]] = 1;

    // ---- layer 1 (delta form) ----
    for (int i = tid; i < 32 * 128; i += 256) {
        int row = i >> 7, c = i & 127;
        float v = 0.f;
        if (row < 24) {
            int t = s_ids[row];
            const float* f = sfeat + ((size_t)b * 24 + row) * 4;
            v = base_h_s[t * 128 + c];
            v = fmaf(f[0], Ws1[128 * 128 + c], v);
            v = fmaf(f[1], Ws1[129 * 128 + c], v);
            v = fmaf(f[2], Ws1[130 * 128 + c], v);
            v = fmaf(f[3], Ws1[131 * 128 + c], v);
            v = fmaxf(v, 0.f);
        }
        h_s[row * 132 + c] = v;
    }
    for (int i = tid; i < 48 * 128; i += 256) {
        int row = i >> 7, c = i & 127;
        float v = 0.f;
        if (row < 40) {
            int t = m_ids[row];
            const float* f = mfeat + ((size_t)b * 40 + row) * 2;
            v = base_h_m[t * 128 + c];
            v = fmaf(f[0], Wm1[128 * 128 + c], v);
            v = fmaf(f[1], Wm1[129 * 128 + c], v);
            v = fmaxf(v, 0.f);
        }
        h_m[row * 132 + c] = v;
    }
    __syncthreads();

    // ---- layer 2: WMMA f32 16x16x4. wave w owns output columns [16w,16w+16) ----
    const int lane = tid & 31;
    const int wv   = tid >> 5;
    const int ln15 = lane & 15;
    const int koff = (lane < 16) ? 0 : 2;   // A/B K sub-slot per half-wave
    const int col  = wv * 16 + ln15;

    // screen MLP second layer: 2 M-tiles (24 active rows padded to 32)
    {
        v2f bw[32];
        #pragma unroll
        for (int kk = 0; kk < 32; ++kk) {
            int k = 4 * kk + koff;
            bw[kk].x = Ws2[k * 128 + col];
            bw[kk].y = Ws2[k * 128 + 128 + col];
        }
        float bias = bs2[col];
        #pragma unroll
        for (int mt = 0; mt < 2; ++mt) {
            v8f acc = {};
            #pragma unroll
            for (int kk = 0; kk < 32; ++kk) {
                v2f a = *(const v2f*)(h_s + (mt * 16 + ln15) * 132 + 4 * kk + koff);
                acc = __builtin_amdgcn_wmma_f32_16x16x4_f32(
                        false, a, false, bw[kk], (short)0, acc, false, false);
            }
            #pragma unroll
            for (int r = 0; r < 8; ++r) {
                int rl = mt * 16 + ((lane < 16) ? r : r + 8);
                if (rl < 24)
                    out[obase + 32 + (size_t)s_ids[rl] * 128 + col] = acc[r] + bias;
            }
        }
    }
    // minimap MLP second layer: 3 M-tiles (40 active rows padded to 48)
    {
        v2f bw[32];
        #pragma unroll
        for (int kk = 0; kk < 32; ++kk) {
            int k = 4 * kk + koff;
            bw[kk].x = Wm2[k * 128 + col];
            bw[kk].y = Wm2[k * 128 + 128 + col];
        }
        float bias = bm2[col];
        #pragma unroll
        for (int mt = 0; mt < 3; ++mt) {
            v8f acc = {};
            #pragma unroll
            for (int kk = 0; kk < 32; ++kk) {
                v2f a = *(const v2f*)(h_m + (mt * 16 + ln15) * 132 + 4 * kk + koff);
                acc = __builtin_amdgcn_wmma_f32_16x16x4_f32(
                        false, a, false, bw[kk], (short)0, acc, false, false);
            }
            #pragma unroll
            for (int r = 0; r < 8; ++r) {
                int rl = mt * 16 + ((lane < 16) ? r : r + 8);
                if (rl < 40)
                    out[obase + 8480 + (size_t)m_ids[rl] * 128 + col] = acc[r] + bias;
            }
        }
    }

    // ---- epilogue: constant rows, item gather, continuous prefix ----
    if (tid < 32) out[obase + tid] = cont[(size_t)b * 32 + tid];
    for (int i = tid; i < TT * 128; i += 256) {
        int t = i >> 7;
        if (!act_s[t]) out[obase + 32 + i]   = def_out_s[i];
        if (!act_m[t]) out[obase + 8480 + i] = def_out_m[i];
    }
    for (int i = tid; i < 70 * 64; i += 256) {
        int j = i >> 6, k = i & 63;
        out[obase + 16928 + i] = item_emb[(size_t)items[(size_t)b * 70 + j] * 64 + k];
    }
}

// ---------------------------------------------------------------------------
extern "C" void kernel_launch(void* const* d_in, const int* in_sizes, int n_in,
                              void* d_out, int out_size, void* d_ws, size_t ws_size,
                              hipStream_t stream) {
    const float* cont     = (const float*)d_in[0];
    const float* sfeat    = (const float*)d_in[1];
    const float* mfeat    = (const float*)d_in[2];
    const float* char_emb = (const float*)d_in[3];
    const float* item_emb = (const float*)d_in[4];
    const float* Ws1      = (const float*)d_in[5];
    const float* bs1      = (const float*)d_in[6];
    const float* Ws2      = (const float*)d_in[7];
    const float* bs2      = (const float*)d_in[8];
    const float* Wm1      = (const float*)d_in[9];
    const float* bm1      = (const float*)d_in[10];
    const float* Wm2      = (const float*)d_in[11];
    const float* bm2      = (const float*)d_in[12];
    const int*   sids     = (const int*)d_in[13];
    const int*   mids     = (const int*)d_in[14];
    const int*   items    = (const int*)d_in[15];
    float*       out      = (float*)d_out;

    const int B = in_sizes[0] / 32;

    float* ws        = (float*)d_ws;
    float* base_h_s  = ws;                 // 66*128
    float* base_h_m  = ws + 1 * TT * 128;
    float* def_out_s = ws + 2 * TT * 128;
    float* def_out_m = ws + 3 * TT * 128;

    const int pre_blocks = (2 * TT * 128 + 255) / 256;
    garen_base_kernel<<<pre_blocks, 256, 0, stream>>>(
        char_emb, Ws1, bs1, Wm1, bm1, base_h_s, base_h_m);
    garen_def_kernel<<<pre_blocks, 256, 0, stream>>>(
        base_h_s, base_h_m, Ws2, bs2, Wm2, bm2, def_out_s, def_out_m);
    garen_main_kernel<<<B, 256, 0, stream>>>(
        cont, sfeat, mfeat, item_emb, Ws1, Ws2, bs2, Wm1, Wm2, bm2,
        sids, mids, items, base_h_s, base_h_m, def_out_s, def_out_m, out);
}